// MultiHeadAttention_5171140624566
// MI455X (gfx1250) — compile-verified
//
#include <hip/hip_runtime.h>
#include <math.h>

// ---- problem constants (match reference) ----
#define NB     8
#define LSEQ   1024
#define DMODEL 1024
#define NH     16
#define DHEAD  64

typedef __attribute__((ext_vector_type(2))) float v2f;
typedef __attribute__((ext_vector_type(4))) float v4f;
typedef __attribute__((ext_vector_type(8))) float v8f;
typedef __attribute__((ext_vector_type(4))) int   v4i;

#define ASG __attribute__((address_space(1)))   // global
#define ASL __attribute__((address_space(3)))   // LDS

#if defined(__has_builtin)
#if __has_builtin(__builtin_amdgcn_global_load_async_to_lds_b128) && \
    __has_builtin(__builtin_amdgcn_global_store_async_from_lds_b128)
#define HAVE_ASYNC_LDS 1
#endif
#endif
#ifndef HAVE_ASYNC_LDS
#define HAVE_ASYNC_LDS 0
#endif

// 16B global -> LDS (async DMA, ASYNCcnt). Builtin signature (from probe):
//   (int4 AS1* gsrc, int4 AS3* lds, imm offset, imm cpol)
__device__ __forceinline__ void stage_g2l_b128(float* lds_dst, const float* gsrc) {
#if HAVE_ASYNC_LDS
  ASG v4i* g = (ASG v4i*)(size_t)gsrc;                      // drops const; AS1
  ASL v4i* l = (ASL v4i*)(unsigned int)(size_t)lds_dst;     // low 32b = LDS offs
  __builtin_amdgcn_global_load_async_to_lds_b128(g, l, 0, 0);
#else
  *(v4f*)lds_dst = *(const v4f*)gsrc;
#endif
}
// 16B LDS -> global (async DMA, ASYNCcnt)
__device__ __forceinline__ void stage_l2g_b128(float* gdst, float* lds_src) {
#if HAVE_ASYNC_LDS
  ASG v4i* g = (ASG v4i*)(size_t)gdst;
  ASL v4i* l = (ASL v4i*)(unsigned int)(size_t)lds_src;
  __builtin_amdgcn_global_store_async_from_lds_b128(g, l, 0, 0);
#else
  *(v4f*)gdst = *(const v4f*)lds_src;
#endif
}
__device__ __forceinline__ void wait_async0() {
#if HAVE_ASYNC_LDS
#if __has_builtin(__builtin_amdgcn_s_wait_asynccnt)
  __builtin_amdgcn_s_wait_asynccnt(0);
#else
  asm volatile("s_wait_asynccnt 0x0" ::: "memory");
#endif
#endif
}

// D = A(16x4 f32) * B(4x16 f32) + C(16x16 f32), wave32
__device__ __forceinline__ v8f wmma_f32(v2f a, v2f b, v8f c) {
  return __builtin_amdgcn_wmma_f32_16x16x4_f32(
      /*neg_a=*/false, a, /*neg_b=*/false, b,
      /*c_mod=*/(short)0, c, /*reuse_a=*/false, /*reuse_b=*/false);
}

// =====================================================================
// GEMM:  Y[m][n] = sum_k A[m][k] * W[n][k]      (A: MxK, W: NxK row-major)
// block = 256 threads (8 waves, 2x4); block tile 64(M) x 128(N);
// each wave: 32x32 tile as 2x2 WMMA accumulators (4x fragment reuse).
// Tiles staged global->LDS with async b128 DMA (ASYNCcnt).
// mode==1 : QKV scatter; z==0/2 -> [B,H,L,DHEAD]; z==1 (K) -> [B,H,DHEAD,L]
// mode==0 : plain row-major MxN
// =====================================================================
__global__ __launch_bounds__(256)
void gemm_xwt_kernel(const float* __restrict__ A,
                     const float* __restrict__ W0, const float* __restrict__ W1,
                     const float* __restrict__ W2,
                     float* __restrict__ Y0, float* __restrict__ Y1,
                     float* __restrict__ Y2,
                     int Kdim, int N, int mode)
{
  const int z = blockIdx.z;
  const float* W = (z == 0) ? W0 : (z == 1 ? W1 : W2);
  float*       Y = (z == 0) ? Y0 : (z == 1 ? Y1 : Y2);

  __shared__ float As[64][68];    // pitch 68: bank-conflict-free
  __shared__ float Ws[128][68];   // total 52.2 KB

  const int tid  = threadIdx.x;
  const int lane = tid & 31;
  const int wave = tid >> 5;
  const int lo   = lane & 15;     // A: M-row / B,C: N-col
  const int hi   = lane >> 4;     // K-half for A/B; M+8 for C
  const int wm   = wave >> 2;     // 0..1 : wave M position
  const int wn   = wave & 3;      // 0..3 : wave N position

  const int m0 = blockIdx.x * 64;
  const int n0 = blockIdx.y * 128;

  v8f acc[2][2];
#pragma unroll
  for (int mi = 0; mi < 2; ++mi)
#pragma unroll
    for (int ni = 0; ni < 2; ++ni)
      acc[mi][ni] = (v8f){0.f,0.f,0.f,0.f,0.f,0.f,0.f,0.f};

  for (int kc = 0; kc < Kdim; kc += 64) {
    __syncthreads();   // previous chunk's LDS reads complete
    // async stage: A 64x64 (1024 x b128), W 128x64 (2048 x b128)
    for (int t = tid; t < 64 * 16; t += 256) {
      int r = t >> 4, c4 = (t & 15) << 2;
      stage_g2l_b128(&As[r][c4], A + (size_t)(m0 + r) * Kdim + kc + c4);
    }
    for (int t = tid; t < 128 * 16; t += 256) {
      int r = t >> 4, c4 = (t & 15) << 2;
      stage_g2l_b128(&Ws[r][c4], W + (size_t)(n0 + r) * Kdim + kc + c4);
    }
    wait_async0();     // this wave's DMAs landed in LDS
    __syncthreads();   // all waves' tiles visible

    // prefetch next K-chunk into L2 while this one computes
    if (kc + 64 < Kdim) {
      __builtin_prefetch(A + (size_t)(m0 + (tid & 63)) * Kdim + kc + 64, 0, 1);
      __builtin_prefetch(W + (size_t)(n0 + (tid & 127)) * Kdim + kc + 64, 0, 1);
    }

#pragma unroll
    for (int t = 0; t < 16; ++t) {
      const int ko = t * 4 + 2 * hi;
      v2f a0 = *(const v2f*)&As[wm * 32 +      lo][ko];
      v2f a1 = *(const v2f*)&As[wm * 32 + 16 + lo][ko];
      v2f b0 = *(const v2f*)&Ws[wn * 32 +      lo][ko];
      v2f b1 = *(const v2f*)&Ws[wn * 32 + 16 + lo][ko];
      acc[0][0] = wmma_f32(a0, b0, acc[0][0]);
      acc[0][1] = wmma_f32(a0, b1, acc[0][1]);
      acc[1][0] = wmma_f32(a1, b0, acc[1][0]);
      acc[1][1] = wmma_f32(a1, b1, acc[1][1]);
    }
  }

#pragma unroll
  for (int mi = 0; mi < 2; ++mi) {
#pragma unroll
    for (int ni = 0; ni < 2; ++ni) {
      const int ncol = n0 + wn * 32 + ni * 16 + lo;
#pragma unroll
      for (int i = 0; i < 8; ++i) {
        const int mrow = m0 + wm * 32 + mi * 16 + i + 8 * hi;
        size_t idx;
        if (mode == 1) {
          int b = mrow >> 10, l = mrow & 1023;
          int h = ncol >> 6,  kk = ncol & 63;
          if (z == 1)  // K stored transposed [B,H,DHEAD,L] for coalesced QK^T
            idx = (((size_t)b * NH + h) * DHEAD + kk) * LSEQ + l;
          else
            idx = (((size_t)b * NH + h) * LSEQ + l) * DHEAD + kk;
        } else {
          idx = (size_t)mrow * N + ncol;
        }
        Y[idx] = acc[mi][ni][i];
      }
    }
  }
}

// =====================================================================
// Attention: one wave per (b, h, 16-query tile).
//   phase 1: S = scale*Q·Kt + mask  -> LDS (16 x 1024), track row max
//   phase 2: exact softmax, normalized IN LDS (inv=0 on empty rows)
//   phase 3: async LDS->global attn write; P·V -> cat workspace
// Kt is [B,H,DHEAD,L] so phase-1 B-frag loads are lane-coalesced.
// =====================================================================
__global__ __launch_bounds__(32)
void attn_kernel(const float* __restrict__ Q, const float* __restrict__ Kt,
                 const float* __restrict__ V, const unsigned char* __restrict__ pad,
                 float* __restrict__ attn, float* __restrict__ cat)
{
  __shared__ float Sbuf[16 * 1024];     // 64 KB: full score rows for 16 queries

  const int lane = threadIdx.x & 31;
  const int lo   = lane & 15;
  const int hi   = lane >> 4;
  const int q0   = blockIdx.x * 16;
  const int bh   = blockIdx.y;
  const int b    = bh >> 4;
  const int h    = bh & 15;
  const float scale = 0.125f;           // 1/sqrt(64)

  const float* Qp = Q  + (size_t)bh * LSEQ * DHEAD;
  const float* Kp = Kt + (size_t)bh * DHEAD * LSEQ;   // [DHEAD][LSEQ]
  const float* Vp = V  + (size_t)bh * LSEQ * DHEAD;
  float* attnp    = attn + (size_t)bh * LSEQ * LSEQ;

  // ---- Q tile (16x64) as register-resident A fragments ----
  v2f qa[16];
#pragma unroll
  for (int t = 0; t < 16; ++t)
    qa[t] = *(const v2f*)(Qp + (size_t)(q0 + lo) * DHEAD + t * 4 + 2 * hi);

  // query-pad additive -inf, distributed per accumulator row (M = i + 8*hi)
  float qn = 0.f;
  if (lane < 16) qn = pad[b * LSEQ + q0 + lane] ? -INFINITY : 0.f;
  float qneg[8];
#pragma unroll
  for (int i = 0; i < 8; ++i) qneg[i] = __shfl(qn, i + 8 * hi, 32);

  float vmax[8];
#pragma unroll
  for (int i = 0; i < 8; ++i) vmax[i] = -INFINITY;

  // ---- phase 1: scores over all key tiles ----
  for (int st = 0; st < 64; ++st) {
    const int col = st * 16 + lo;                 // global key index == C col
    const float kneg = pad[b * LSEQ + col] ? -INFINITY : 0.f;
    v8f s = {0.f,0.f,0.f,0.f,0.f,0.f,0.f,0.f};
#pragma unroll
    for (int t = 0; t < 16; ++t) {
      const int k = t * 4 + 2 * hi;
      v2f bf;     // b[j] = K[key=col][k+j] = Kt[k+j][col] (lane-coalesced)
      bf.x = Kp[(size_t)k       * LSEQ + col];
      bf.y = Kp[(size_t)(k + 1) * LSEQ + col];
      s = wmma_f32(qa[t], bf, s);
    }
#pragma unroll
    for (int i = 0; i < 8; ++i) {
      float val = s[i] * scale + qneg[i] + kneg;
      Sbuf[(i + 8 * hi) * 1024 + col] = val;
      vmax[i] = fmaxf(vmax[i], val);
    }
  }
  __syncthreads();

  // ---- row max across the 16 lanes that share a row ----
#pragma unroll
  for (int i = 0; i < 8; ++i) {
#pragma unroll
    for (int off = 1; off < 16; off <<= 1)
      vmax[i] = fmaxf(vmax[i], __shfl_xor(vmax[i], off, 32));
    if (vmax[i] < -1e37f) vmax[i] = 0.f;          // fully-masked row guard
  }

  // ---- phase 2a: exp sweep + row sums (float4 in-place) ----
  float rsum[8] = {0.f,0.f,0.f,0.f,0.f,0.f,0.f,0.f};
  for (int st = 0; st < 16; ++st) {
#pragma unroll
    for (int i = 0; i < 8; ++i) {
      int idx = (i + 8 * hi) * 1024 + st * 64 + lo * 4;
      v4f p = *(const v4f*)&Sbuf[idx];
      v4f e;
      e.x = __expf(p.x - vmax[i]);  e.y = __expf(p.y - vmax[i]);
      e.z = __expf(p.z - vmax[i]);  e.w = __expf(p.w - vmax[i]);
      *(v4f*)&Sbuf[idx] = e;
      rsum[i] += e.x + e.y + e.z + e.w;
    }
  }
  float inv[8];
#pragma unroll
  for (int i = 0; i < 8; ++i) {
#pragma unroll
    for (int off = 1; off < 16; off <<= 1)
      rsum[i] += __shfl_xor(rsum[i], off, 32);
    inv[i] = (rsum[i] > 0.f) ? (1.f / rsum[i]) : 0.f;   // == nan_to_num
  }

  // ---- phase 2b: normalize in LDS (P = exp * inv) ----
  for (int st = 0; st < 16; ++st) {
#pragma unroll
    for (int i = 0; i < 8; ++i) {
      int idx = (i + 8 * hi) * 1024 + st * 64 + lo * 4;
      v4f p = *(const v4f*)&Sbuf[idx];
      p.x *= inv[i]; p.y *= inv[i]; p.z *= inv[i]; p.w *= inv[i];
      *(v4f*)&Sbuf[idx] = p;
    }
  }
  __syncthreads();   // order LDS writes before async LDS->global DMA

  // ---- phase 2c: async DMA the normalized 64KB tile to global attn ----
  for (int st = 0; st < 16; ++st) {
#pragma unroll
    for (int i = 0; i < 8; ++i) {
      int r = i + 8 * hi, c = st * 64 + lo * 4;
      stage_l2g_b128(attnp + (size_t)(q0 + r) * LSEQ + c, &Sbuf[r * 1024 + c]);
    }
  }

  // ---- phase 3: out = P · V  (16 x 64 tile, 4 accumulators) ----
  v8f acc[4];
#pragma unroll
  for (int v = 0; v < 4; ++v) acc[v] = (v8f){0.f,0.f,0.f,0.f,0.f,0.f,0.f,0.f};

  for (int st = 0; st < 64; ++st) {
#pragma unroll
    for (int c = 0; c < 4; ++c) {
      const int s0 = st * 16 + c * 4;
      // A frag: a[j] = P[q=lo][s = s0 + 2*hi + j] (already normalized)
      v2f a = *(const v2f*)&Sbuf[lo * 1024 + s0 + 2 * hi];
#pragma unroll
      for (int v = 0; v < 4; ++v) {
        v2f bf;   // b[j] = V[s = s0 + 2*hi + j][col = v*16 + lo] (coalesced)
        bf.x = Vp[(size_t)(s0 + 2 * hi)     * DHEAD + v * 16 + lo];
        bf.y = Vp[(size_t)(s0 + 2 * hi + 1) * DHEAD + v * 16 + lo];
        acc[v] = wmma_f32(a, bf, acc[v]);
      }
    }
  }

  // store out_h into cat workspace [B, L, H*DV] (head-concat layout)
#pragma unroll
  for (int v = 0; v < 4; ++v) {
#pragma unroll
    for (int i = 0; i < 8; ++i) {
      int r = q0 + i + 8 * hi;
      cat[((size_t)b * LSEQ + r) * DMODEL + h * DHEAD + v * 16 + lo] = acc[v][i];
    }
  }

  wait_async0();   // drain attn DMA before wave end (S_ENDPGM also waits)
}

// =====================================================================
extern "C" void kernel_launch(void* const* d_in, const int* in_sizes, int n_in,
                              void* d_out, int out_size, void* d_ws, size_t ws_size,
                              hipStream_t stream) {
  const float* X  = (const float*)d_in[0];           // [B,L,D]
  const float* WQ = (const float*)d_in[1];           // [H,DK,D]
  const float* WK = (const float*)d_in[2];
  const float* WV = (const float*)d_in[3];
  const float* WO = (const float*)d_in[4];           // [D,H*DV]
  const unsigned char* pad = (const unsigned char*)d_in[5];  // [B,L] bool

  float* out  = (float*)d_out;                                   // [B,L,D]
  float* attn = out + (size_t)NB * LSEQ * DMODEL;                // [B,H,L,L]

  const size_t qkv_elems = (size_t)NB * NH * LSEQ * DHEAD;       // 8.39M each
  float* Q   = (float*)d_ws;                                     // [B,H,L,DK]
  float* Kt  = Q + qkv_elems;                                    // [B,H,DK,L]
  float* V   = Kt + qkv_elems;                                   // [B,H,L,DV]
  float* cat = V + qkv_elems;                                    // [B,L,H*DV]

  // 1) Q/K/V projections: M=B*L=8192, N=H*DK=1024, K=D=1024 (z selects W)
  dim3 g1((NB * LSEQ) / 64, (NH * DHEAD) / 128, 3);
  gemm_xwt_kernel<<<g1, 256, 0, stream>>>(X, WQ, WK, WV, Q, Kt, V,
                                          DMODEL, NH * DHEAD, /*mode=*/1);

  // 2) attention: one wave per (16-query tile, b*h)
  dim3 g2(LSEQ / 16, NB * NH, 1);
  attn_kernel<<<g2, 32, 0, stream>>>(Q, Kt, V, pad, attn, cat);

  // 3) output projection: out = cat @ WO^T  (M=8192, N=1024, K=1024)
  dim3 g3((NB * LSEQ) / 64, DMODEL / 128, 1);
  gemm_xwt_kernel<<<g3, 256, 0, stream>>>(cat, WO, WO, WO, out, out, out,
                                          DMODEL, DMODEL, /*mode=*/0);
}